// PredictionDecoder_80023830659235
// MI455X (gfx1250) — compile-verified
//
#include <hip/hip_runtime.h>
#include <hip/hip_bf16.h>
#include <stdint.h>

#ifndef __has_builtin
#define __has_builtin(x) 0
#endif

#define B_      4
#define N_      49104
#define C_      2
#define K_      550            // MAX_PER_CLASS
#define TOTAL_  (C_ * K_)      // 1100 == MAX_TOTAL
#define NEG_    (-1000000000.0f)
#define CONF_   0.05f
#define IOUT_   0.15f

#define NMS_THREADS 1024
#define EPT ((N_ + NMS_THREADS - 1) / NMS_THREADS)   // 48 elements per thread

// ---------------------------------------------------------------------------
// Kernel 1: decode cxcywh -> corners, sigmoid+threshold logits -> score array
// ---------------------------------------------------------------------------
__global__ void decode_corners_scores(const float* __restrict__ pred,   // [B,N,4+C]
                                      float* __restrict__ corners,      // [B*N,4]
                                      float* __restrict__ scores)       // [B*C,N]
{
    int g = blockIdx.x * blockDim.x + threadIdx.x;
    if (g >= B_ * N_) return;
    const float* p = pred + (size_t)g * (4 + C_);
    float cx = p[0], cy = p[1], w = p[2], h = p[3];
    float4 cr;
    cr.x = cx - w * 0.5f;
    cr.y = cy - h * 0.5f;
    cr.z = cx + w * 0.5f;
    cr.w = cy + h * 0.5f;
    ((float4*)corners)[g] = cr;
    int b = g / N_;
    int i = g - b * N_;
#pragma unroll
    for (int c = 0; c < C_; ++c) {
        float x = p[4 + c];
        float s = 1.0f / (1.0f + expf(-x));               // jax.nn.sigmoid
        scores[((size_t)(b * C_ + c)) * N_ + i] = (s > CONF_) ? s : NEG_;
    }
}

// ---------------------------------------------------------------------------
// order-preserving float <-> uint key helpers (max key == max score, min idx)
// ---------------------------------------------------------------------------
__device__ __forceinline__ unsigned int flip_f32(float f) {
    unsigned int u = __float_as_uint(f);
    return (u & 0x80000000u) ? ~u : (u | 0x80000000u);
}
__device__ __forceinline__ float unflip_f32(unsigned int k) {
    unsigned int u = (k & 0x80000000u) ? (k & 0x7fffffffu) : ~k;
    return __uint_as_float(u);
}

typedef unsigned int u32x4_t __attribute__((ext_vector_type(4)));
typedef int          i32x8_t __attribute__((ext_vector_type(8)));
typedef int          i32x4_t __attribute__((ext_vector_type(4)));

#if __has_builtin(__builtin_amdgcn_global_load_async_to_lds_b32) && __has_builtin(__builtin_amdgcn_s_wait_asynccnt)
#define HAVE_ASYNC_LDS 1
typedef __attribute__((address_space(1))) int* as1_int_p;   // global ("__device__")
typedef __attribute__((address_space(3))) int* as3_int_p;   // LDS
#else
#define HAVE_ASYNC_LDS 0
#endif

// ---------------------------------------------------------------------------
// Kernel 2: serial greedy NMS, one workgroup per (batch, class).
// Scores live in LDS (192KB of the CDNA5 320KB WGP LDS), staged via the
// Tensor Data Mover (tensor_load_to_lds + s_wait_tensorcnt).
// ---------------------------------------------------------------------------
__global__ void __launch_bounds__(NMS_THREADS)
nms_kernel(const float* __restrict__ corners,   // [B*N,4]
           const float* __restrict__ scores,    // [B*C,N]
           float* __restrict__ selScore,        // [B*C,K]
           float* __restrict__ selBox)          // [B*C,K,4]
{
    extern __shared__ float sc_lds[];                    // N_ floats (dynamic)
    __shared__ unsigned long long warr[NMS_THREADS / 32];
    __shared__ float4 s_box;
    __shared__ unsigned int s_idx;
    __shared__ int s_stop;

    const int bc  = blockIdx.x;
    const int b   = bc / C_;
    const int tid = threadIdx.x;

    const float4* __restrict__ cb = (const float4*)corners + (size_t)b * N_;
    const float*  __restrict__ sg = scores + (size_t)bc * N_;
    float*  outS = selScore + (size_t)bc * K_;
    float4* outB = (float4*)selBox + (size_t)bc * K_;

    // ---- stage per-(b,c) score array global -> LDS via TDM -----------------
    if (tid < 32) {   // one wave issues the tensor DMA
        uint64_t ga      = (uint64_t)(uintptr_t)sg;
        uint32_t lds_off = (uint32_t)(size_t)(void*)sc_lds;  // low 32 bits of flat addr = LDS offset
        u32x4_t g0;
        g0[0] = 1u;                                          // count = 1 (user descriptor)
        g0[1] = lds_off;                                     // lds_addr
        g0[2] = (uint32_t)ga;                                // global_addr[31:0]
        g0[3] = (uint32_t)((ga >> 32) & 0x01ffffffu) | (2u << 30); // addr[56:32] | type=2
        i32x8_t g1;
        g1[0] = (int)(2u << 16);                             // data_size = 4 bytes
        g1[1] = (int)(((uint32_t)N_ & 0xffffu) << 16);       // tensor_dim0[15:0]
        g1[2] = (int)(((uint32_t)N_ >> 16) | (1u << 16));    // tensor_dim0[31:16], tensor_dim1=1
        g1[3] = (int)(((uint32_t)N_ & 0xffffu) << 16);       // tile_dim0 = N_ (< 65536)
        g1[4] = 1;                                           // tile_dim1 = 1, tile_dim2 = 0
        g1[5] = (int)(uint32_t)N_;                           // tensor_dim0_stride lo32
        g1[6] = 0;
        g1[7] = 0;
        i32x4_t gz = {0, 0, 0, 0};
#if __clang_major__ >= 23
        i32x8_t gz8 = {0, 0, 0, 0, 0, 0, 0, 0};
        __builtin_amdgcn_tensor_load_to_lds(g0, g1, gz, gz, gz8, 0);
#else
        __builtin_amdgcn_tensor_load_to_lds(g0, g1, gz, gz, 0);
#endif
        __builtin_amdgcn_s_wait_tensorcnt(0);
    }

    // warm L2 with this batch's corner boxes (global_prefetch_b8)
#pragma unroll 1
    for (int j = 0; j < EPT; ++j) {
        int i = tid + j * NMS_THREADS;
        if (i < N_) __builtin_prefetch(&cb[i], 0, 0);
    }
    __syncthreads();

    int fill_from = K_;
#pragma unroll 1
    for (int k = 0; k < K_; ++k) {
        // ---- block-wide argmax (first-index tie-break, like jnp.argmax) ----
        unsigned long long best = 0ull;
#pragma unroll 4
        for (int j = 0; j < EPT; ++j) {
            int i = tid + j * NMS_THREADS;
            if (i < N_) {
                unsigned long long key =
                    ((unsigned long long)flip_f32(sc_lds[i]) << 32) |
                    (unsigned int)(~(unsigned int)i);
                if (key > best) best = key;
            }
        }
        for (int off = 16; off > 0; off >>= 1) {
            unsigned long long o = __shfl_xor(best, off, 32);
            if (o > best) best = o;
        }
        const int lane = tid & 31, wid = tid >> 5;
        if (lane == 0) warr[wid] = best;
        __syncthreads();
        if (wid == 0) {
            best = warr[lane];
            for (int off = 16; off > 0; off >>= 1) {
                unsigned long long o = __shfl_xor(best, off, 32);
                if (o > best) best = o;
            }
            if (lane == 0) {
                unsigned int idx = ~(unsigned int)best;
                float sc = unflip_f32((unsigned int)(best >> 32));
                s_idx = idx;
                float4 bb = cb[idx];
                s_box = bb;
                outS[k] = sc;
                outB[k] = bb;
                s_stop = (sc <= 0.5f * NEG_) ? 1 : 0;  // pool exhausted
            }
        }
        __syncthreads();
        if (s_stop) { fill_from = k + 1; break; }

        // ---- suppression: IoU(best, all) > 0.15  ->  NEG -------------------
        const float4 bb = s_box;
        const unsigned int bi = s_idx;
        const float aw = fmaxf(bb.z - bb.x, 0.0f);
        const float ah = fmaxf(bb.w - bb.y, 0.0f);
        const float areaA = aw * ah;
#pragma unroll 4
        for (int j = 0; j < EPT; ++j) {
            int i = tid + j * NMS_THREADS;
            if (i < N_) {
                float s = sc_lds[i];
                if (s != NEG_) {   // NEG entries stay NEG either way
                    float4 c = cb[i];
                    float bw   = fmaxf(c.z - c.x, 0.0f);
                    float bh   = fmaxf(c.w - c.y, 0.0f);
                    float ltx  = fmaxf(bb.x, c.x);
                    float lty  = fmaxf(bb.y, c.y);
                    float rbx  = fminf(bb.z, c.z);
                    float rby  = fminf(bb.w, c.w);
                    float iw   = fmaxf(rbx - ltx, 0.0f);
                    float ih   = fmaxf(rby - lty, 0.0f);
                    float inter = iw * ih;
                    float uni   = areaA + bw * bh - inter;
                    float iou   = inter / fmaxf(uni, 1e-8f);
                    if (iou > IOUT_ || (unsigned int)i == bi) sc_lds[i] = NEG_;
                }
            }
        }
        __syncthreads();
    }

    // tail after early exit: reference keeps emitting (idx 0, NEG); those are
    // invalid (score <= 0.5*NEG) and get zeroed downstream, so only the NEG
    // score matters for the final sort.
    for (int k = fill_from + tid; k < K_; k += NMS_THREADS) {
        outS[k] = NEG_;
        outB[k] = make_float4(0.f, 0.f, 0.f, 0.f);
    }
}

// ---------------------------------------------------------------------------
// Kernel 3: per-batch stable descending sort of 1100 entries (== lax.top_k),
// gather + validity masking + valid count. Scores staged into LDS via the
// gfx1250 async global->LDS path.
// ---------------------------------------------------------------------------
__global__ void __launch_bounds__(1024)
topk_kernel(const float* __restrict__ selScore,   // [B, C*K]
            const float* __restrict__ selBox,     // [B, C*K, 4]
            float* __restrict__ out)
{
    __shared__ float ss[TOTAL_];
    __shared__ int cnt;
    const int b   = blockIdx.x;
    const int tid = threadIdx.x;
    const float*  sS = selScore + (size_t)b * TOTAL_;
    const float4* sB = (const float4*)selBox + (size_t)b * TOTAL_;

#if HAVE_ASYNC_LDS
    for (int j = tid; j < TOTAL_; j += blockDim.x) {
        __builtin_amdgcn_global_load_async_to_lds_b32(
            (as1_int_p)(sS + j),     // global src  (addrspace(1) int*)
            (as3_int_p)(ss + j),     // LDS dst     (addrspace(3) int*)
            0, 0);
    }
    __builtin_amdgcn_s_wait_asynccnt(0);
#else
    for (int j = tid; j < TOTAL_; j += blockDim.x) ss[j] = sS[j];
#endif
    if (tid == 0) cnt = 0;
    __syncthreads();

    float4* outBoxes  = (float4*)out;                         // [B,1100,4]
    float* outScores  = out + (size_t)B_ * TOTAL_ * 4;        // [B,1100]
    float* outClasses = outScores + (size_t)B_ * TOTAL_;      // [B,1100]
    int*   outValid   = (int*)out + (size_t)B_ * TOTAL_ * 6;  // [B] int32

    int local = 0;
    for (int j = tid; j < TOTAL_; j += blockDim.x) {
        float sj = ss[j];
        int rank = 0;
        for (int m = 0; m < TOTAL_; ++m) {
            float sm = ss[m];
            rank += (int)((sm > sj) || ((sm == sj) && (m < j)));
        }
        bool valid = sj > 0.5f * NEG_;
        float4 bx = valid ? sB[j] : make_float4(0.f, 0.f, 0.f, 0.f);
        int pos = b * TOTAL_ + rank;
        outBoxes[pos]   = bx;
        outScores[pos]  = valid ? sj : 0.0f;
        outClasses[pos] = valid ? (float)(j / K_) : 0.0f;
        if (valid) ++local;
    }
    if (local) atomicAdd(&cnt, local);
    __syncthreads();
    if (tid == 0) outValid[b] = cnt;
}

// ---------------------------------------------------------------------------
extern "C" void kernel_launch(void* const* d_in, const int* in_sizes, int n_in,
                              void* d_out, int out_size, void* d_ws, size_t ws_size,
                              hipStream_t stream) {
    (void)in_sizes; (void)n_in; (void)out_size; (void)ws_size;
    const float* pred = (const float*)d_in[1];   // d_in[0] = images (unused)
    float* ws      = (float*)d_ws;
    float* corners = ws;                                   // B*N*4   floats
    float* scores  = corners + (size_t)B_ * N_ * 4;        // B*C*N   floats
    float* selS    = scores  + (size_t)B_ * C_ * N_;       // B*C*K   floats
    float* selB    = selS    + (size_t)B_ * C_ * K_;       // B*C*K*4 floats

    int total = B_ * N_;
    decode_corners_scores<<<(total + 255) / 256, 256, 0, stream>>>(pred, corners, scores);

    size_t smem = (size_t)N_ * sizeof(float);              // 192KB of the 320KB WGP LDS
    (void)hipFuncSetAttribute((const void*)nms_kernel,
                              hipFuncAttributeMaxDynamicSharedMemorySize, (int)smem);
    nms_kernel<<<B_ * C_, NMS_THREADS, smem, stream>>>(corners, scores, selS, selB);

    topk_kernel<<<B_, 1024, 0, stream>>>(selS, selB, (float*)d_out);
}